// GptOssDecoderLayer_27857157882046
// MI455X (gfx1250) — compile-verified
//
#include <hip/hip_runtime.h>
#include <hip/hip_bf16.h>

// ---- problem constants ----
#define S_   128
#define H_   512
#define NH_  8
#define NKV_ 2
#define HD_  64
#define E_   32
#define KTOP_ 4
#define F_   256
#define QKVN_ 768   // NH*HD + NKV*HD + NKV*HD

__constant__ constexpr float EPS_f   = 1e-6f;
__constant__ constexpr float ALPHA_f = 1.702f;
__constant__ constexpr float LIMIT_f = 7.0f;
__constant__ constexpr float THETA_f = 150000.0f;

typedef __attribute__((ext_vector_type(2))) float v2f;
typedef __attribute__((ext_vector_type(8))) float v8f;

// D = A(16x4,f32) * B(4x16,f32) + C(16x16,f32)  — CDNA5 V_WMMA_F32_16X16X4_F32
__device__ __forceinline__ v8f wmma4(v2f a, v2f b, v8f c) {
  return __builtin_amdgcn_wmma_f32_16x16x4_f32(
      /*neg_a=*/false, a, /*neg_b=*/false, b,
      /*c_mod=*/(short)0, c, /*reuse_a=*/false, /*reuse_b=*/false);
}

// Generic 16x16 output tile GEMM step over K, all strides compile-time.
// Arow: A row base for this lane (row = m0 + (lane&15)); already includes row offset.
// Bcol: B column base for this lane (col = n0 + (lane&15)).
template <int KSZ, int LDB>
__device__ __forceinline__ void gemm_tile(const float* __restrict__ Arow,
                                          const float* __restrict__ Bcol,
                                          int kb, v8f& acc) {
#pragma unroll 4
  for (int kk = 0; kk < KSZ; kk += 4) {
    v2f a = *(const v2f*)(Arow + kk + kb);              // 8B-aligned (kb in {0,2})
    v2f b;
    b.x = Bcol[(kk + kb) * LDB];
    b.y = Bcol[(kk + kb + 1) * LDB];
    acc = wmma4(a, b, acc);
  }
}

// ---------------- RMSNorm: out = x * rsqrt(mean(x^2)+eps) * w ----------------
__global__ void rms_kernel(const float* __restrict__ x, const float* __restrict__ w,
                           float* __restrict__ out) {
  int t = blockIdx.x;
  __shared__ float red[8];
  int tid = threadIdx.x;                     // 256 threads = 8 waves
  float s = 0.f;
  for (int i = tid; i < H_; i += 256) { float v = x[t * H_ + i]; s += v * v; }
  for (int off = 16; off >= 1; off >>= 1) s += __shfl_xor(s, off, 32);
  if ((tid & 31) == 0) red[tid >> 5] = s;
  __syncthreads();
  float tot = 0.f;
#pragma unroll
  for (int i = 0; i < 8; i++) tot += red[i];
  float scale = rsqrtf(tot / (float)H_ + EPS_f);
  for (int i = tid; i < H_; i += 256) out[t * H_ + i] = x[t * H_ + i] * scale * w[i];
}

// ---------------- Fused QKV GEMM: h[128x512] @ {q_w|k_w|v_w} + bias ----------------
// grid (8, 48), block 32 (1 wave per 16x16 output tile); constant-stride branches
__global__ void qkv_kernel(const float* __restrict__ h,
                           const float* __restrict__ q_w, const float* __restrict__ q_b,
                           const float* __restrict__ k_w, const float* __restrict__ k_b,
                           const float* __restrict__ v_w, const float* __restrict__ v_b,
                           float* __restrict__ qo, float* __restrict__ ko, float* __restrict__ vo) {
  int mt = blockIdx.x;
  int n0 = blockIdx.y * 16;
  int lane = threadIdx.x;
  int rr = lane & 15;
  int kb = (lane >> 4) << 1;
  int rowbase = mt * 16 + ((lane >> 4) << 3);
  const float* Arow = h + (mt * 16 + rr) * H_;
  v8f acc = {};
  if (n0 < 512) {
    gemm_tile<H_, 512>(Arow, q_w + n0 + rr, kb, acc);
    float bv = q_b[n0 + rr];
    float* o = qo + rowbase * 512 + n0 + rr;
#pragma unroll
    for (int r2 = 0; r2 < 8; r2++) o[r2 * 512] = acc[r2] + bv;
  } else if (n0 < 640) {
    int nl = n0 - 512;
    gemm_tile<H_, 128>(Arow, k_w + nl + rr, kb, acc);
    float bv = k_b[nl + rr];
    float* o = ko + rowbase * 128 + nl + rr;
#pragma unroll
    for (int r2 = 0; r2 < 8; r2++) o[r2 * 128] = acc[r2] + bv;
  } else {
    int nl = n0 - 640;
    gemm_tile<H_, 128>(Arow, v_w + nl + rr, kb, acc);
    float bv = v_b[nl + rr];
    float* o = vo + rowbase * 128 + nl + rr;
#pragma unroll
    for (int r2 = 0; r2 < 8; r2++) o[r2 * 128] = acc[r2] + bv;
  }
}

// ---------------- RoPE in place on q or k ----------------
__global__ void rope_kernel(float* __restrict__ x, const int* __restrict__ pos, int nh) {
  int idx = blockIdx.x * blockDim.x + threadIdx.x;
  const int half = HD_ / 2;
  int total = S_ * nh * half;
  if (idx >= total) return;
  int i = idx % half;
  int hh = (idx / half) % nh;
  int t = idx / (half * nh);
  float inv = powf(THETA_f, -((float)i) / (float)half);
  float ang = (float)pos[t] * inv;
  float sn, cs;
  __sincosf(ang, &sn, &cs);
  float* p = x + (t * nh + hh) * HD_;
  float x1 = p[i], x2 = p[i + half];
  p[i]        = x1 * cs - x2 * sn;
  p[i + half] = x2 * cs + x1 * sn;
}

// ---------------- Attention (per head): QK^T (WMMA), softmax+sink, P@V (WMMA) ----------------
// grid NH, block 128 (4 waves); each wave handles query tiles wv and wv+4
__global__ void attn_kernel(const float* __restrict__ q, const float* __restrict__ k,
                            const float* __restrict__ v, const float* __restrict__ sinks,
                            float* __restrict__ attn) {
  __shared__ float pl[4 * 16 * 132];         // per-wave padded 16x128 prob tiles
  int h = blockIdx.x;
  int lane = threadIdx.x & 31;
  int wv = threadIdx.x >> 5;
  int kvh = h >> 2;                          // rep = NH/NKV = 4
  int rr = lane & 15;
  int kb = (lane >> 4) << 1;
  int hi8 = (lane >> 4) << 3;
  const float* Q  = q + h * HD_;             // row stride NH*HD
  const float* Kp = k + kvh * HD_;           // row stride NKV*HD
  const float* Vp = v + kvh * HD_;
  float sinkv = sinks[h];
  float* P = pl + wv * 16 * 132;

  for (int mt = wv; mt < 8; mt += 4) {
    v8f acc[8] = {};
    const float* Arow = Q + (mt * 16 + rr) * (NH_ * HD_);
#pragma unroll
    for (int nt = 0; nt < 8; nt++) {
      const float* Brow = Kp + (nt * 16 + rr) * (NKV_ * HD_);
#pragma unroll
      for (int kk = 0; kk < HD_; kk += 4) {
        v2f a = *(const v2f*)(Arow + kk + kb);
        v2f b = *(const v2f*)(Brow + kk + kb);   // B[d][key]=K[key][d]: same lane pattern
        acc[nt] = wmma4(a, b, acc[nt]);
      }
    }
    // scale + causal mask
#pragma unroll
    for (int nt = 0; nt < 8; nt++)
#pragma unroll
      for (int r2 = 0; r2 < 8; r2++) {
        int row = mt * 16 + r2 + hi8;
        int col = nt * 16 + rr;
        float sc = acc[nt][r2] * 0.125f;     // 1/sqrt(64)
        acc[nt][r2] = (col <= row) ? sc : -1e30f;
      }
    // row-wise softmax with sink column (reduce across 16 lanes of each half)
#pragma unroll
    for (int r2 = 0; r2 < 8; r2++) {
      float m = sinkv;
#pragma unroll
      for (int nt = 0; nt < 8; nt++) m = fmaxf(m, acc[nt][r2]);
      for (int off = 8; off >= 1; off >>= 1) m = fmaxf(m, __shfl_xor(m, off, 32));
      float sp = 0.f;
#pragma unroll
      for (int nt = 0; nt < 8; nt++) {
        float e = __expf(acc[nt][r2] - m);
        acc[nt][r2] = e;
        sp += e;
      }
      for (int off = 8; off >= 1; off >>= 1) sp += __shfl_xor(sp, off, 32);
      float inv = 1.0f / (sp + __expf(sinkv - m));
#pragma unroll
      for (int nt = 0; nt < 8; nt++) acc[nt][r2] *= inv;
    }
    // stage probs through LDS (D layout -> A layout)
#pragma unroll
    for (int nt = 0; nt < 8; nt++)
#pragma unroll
      for (int r2 = 0; r2 < 8; r2++)
        P[(r2 + hi8) * 132 + nt * 16 + rr] = acc[nt][r2];

    // P(16x128) @ V(128x64)
    v8f oacc[4] = {};
    const float* Prow = P + rr * 132;
#pragma unroll 4
    for (int kk = 0; kk < S_; kk += 4) {
      v2f a = *(const v2f*)(Prow + kk + kb);
#pragma unroll
      for (int nt = 0; nt < 4; nt++) {
        v2f b;
        b.x = Vp[(kk + kb) * (NKV_ * HD_) + nt * 16 + rr];
        b.y = Vp[(kk + kb + 1) * (NKV_ * HD_) + nt * 16 + rr];
        oacc[nt] = wmma4(a, b, oacc[nt]);
      }
    }
    float* Orow = attn + (mt * 16 + hi8) * (NH_ * HD_) + h * HD_ + rr;
#pragma unroll
    for (int nt = 0; nt < 4; nt++)
#pragma unroll
      for (int r2 = 0; r2 < 8; r2++)
        Orow[r2 * (NH_ * HD_) + nt * 16] = oacc[nt][r2];
  }
}

// ---------------- O projection + residual: x2 = hidden + attn @ o_w + o_b ----------------
// grid (8, 32), block 32
__global__ void oproj_kernel(const float* __restrict__ attn, const float* __restrict__ o_w,
                             const float* __restrict__ o_b, const float* __restrict__ xin,
                             float* __restrict__ x2) {
  int mt = blockIdx.x;
  int n0 = blockIdx.y * 16;
  int lane = threadIdx.x;
  int rr = lane & 15;
  int kb = (lane >> 4) << 1;
  v8f acc = {};
  const float* Arow = attn + (mt * 16 + rr) * H_;
  gemm_tile<H_, H_>(Arow, o_w + n0 + rr, kb, acc);
  float bv = o_b[n0 + rr];
  int rowbase = mt * 16 + ((lane >> 4) << 3);
  const float* xi = xin + rowbase * H_ + n0 + rr;
  float* xo = x2 + rowbase * H_ + n0 + rr;
#pragma unroll
  for (int r2 = 0; r2 < 8; r2++) xo[r2 * H_] = xi[r2 * H_] + acc[r2] + bv;
}

// ---------------- Router: logits, top-4, softmax weights ----------------
// grid S, block 32 (lane = expert)
__global__ void router_kernel(const float* __restrict__ h2, const float* __restrict__ rw,
                              const float* __restrict__ rb, int* __restrict__ top_i,
                              float* __restrict__ top_w) {
  int t = blockIdx.x;
  int lane = threadIdx.x;
  const float* hrow = h2 + t * H_;
  const float* wrow = rw + lane * H_;
  float acc = rb[lane];
  for (int d = 0; d < H_; d++) acc = fmaf(hrow[d], wrow[d], acc);
  float val = acc;
  float bv[KTOP_]; int bi[KTOP_];
#pragma unroll
  for (int s2 = 0; s2 < KTOP_; s2++) {
    float m = val; int mi = lane;
    for (int off = 16; off >= 1; off >>= 1) {
      float om = __shfl_xor(m, off, 32);
      int   oi = __shfl_xor(mi, off, 32);
      if (om > m || (om == m && oi < mi)) { m = om; mi = oi; }
    }
    bv[s2] = m; bi[s2] = mi;
    if (lane == mi) val = -1e30f;
  }
  float mx = bv[0];
  float e0 = __expf(bv[0] - mx), e1 = __expf(bv[1] - mx);
  float e2 = __expf(bv[2] - mx), e3 = __expf(bv[3] - mx);
  float inv = 1.0f / (e0 + e1 + e2 + e3);
  float wv4[KTOP_] = {e0 * inv, e1 * inv, e2 * inv, e3 * inv};
  if (lane < KTOP_) {
    top_i[t * KTOP_ + lane] = bi[lane];
    top_w[t * KTOP_ + lane] = wv4[lane];
  }
}

// ---------------- Build per-expert token lists ----------------
// grid 1, block 512 (one thread per (token,slot))
__global__ void scatter_kernel(const int* __restrict__ top_i, int* __restrict__ counts,
                               int* __restrict__ list_tok, int* __restrict__ list_slot) {
  int tid = threadIdx.x;
  if (tid < E_) counts[tid] = 0;
  __syncthreads();
  int t = tid >> 2;
  int e = top_i[tid];
  int p = atomicAdd(&counts[e], 1);
  list_tok[e * S_ + p]  = t;
  list_slot[e * S_ + p] = tid;
}

// ---------------- Grouped gate_up GEMM + SwiGLU ----------------
// grid (E, 8 m-tiles), block 256 (8 waves; wave covers 64 output columns)
__global__ void moe_gu_kernel(const float* __restrict__ h2, const float* __restrict__ gup,
                              const float* __restrict__ gub, const int* __restrict__ counts,
                              const int* __restrict__ list_tok, float* __restrict__ fused) {
  int e = blockIdx.x, mt = blockIdx.y;
  int cnt = counts[e];
  if (mt * 16 >= cnt) return;
  int lane = threadIdx.x & 31, wv = threadIdx.x >> 5;
  int rr = lane & 15, kb = (lane >> 4) << 1, hi8 = (lane >> 4) << 3;
  int p = mt * 16 + rr;
  int tok = list_tok[e * S_ + min(p, cnt - 1)];
  const float* Arow = h2 + tok * H_;
  const float* B = gup + (size_t)e * H_ * (2 * F_);
  const float* bias = gub + e * (2 * F_);
  int n0 = wv * 64;
  v8f acc[4] = {};
#pragma unroll 2
  for (int kk = 0; kk < H_; kk += 4) {
    v2f a = *(const v2f*)(Arow + kk + kb);
    const float* B0 = B + (size_t)(kk + kb) * (2 * F_) + n0 + rr;
    const float* B1 = B0 + (2 * F_);
#pragma unroll
    for (int nt = 0; nt < 4; nt++) {
      v2f b; b.x = B0[nt * 16]; b.y = B1[nt * 16];
      acc[nt] = wmma4(a, b, acc[nt]);
    }
  }
#pragma unroll
  for (int nt = 0; nt < 4; nt++) {
    int col = n0 + nt * 16 + rr;
    float bval = bias[col];
#pragma unroll
    for (int r2 = 0; r2 < 8; r2++) {
      float gu = acc[nt][r2] + bval;
      float other = __shfl_xor(gu, 1, 32);     // pair even<->odd column
      if (!(rr & 1)) {
        float gate = fminf(gu, LIMIT_f);
        float up = fminf(fmaxf(other, -LIMIT_f), LIMIT_f);
        float sw = gate / (1.0f + __expf(-gate * ALPHA_f));
        float fv = (up + 1.0f) * sw;
        int prow = mt * 16 + r2 + hi8;
        if (prow < cnt) fused[(size_t)(e * S_ + prow) * F_ + (col >> 1)] = fv;
      }
    }
  }
}

// ---------------- Grouped down GEMM, scatter to slot buffer ----------------
// grid (E, 8 m-tiles), block 256
__global__ void moe_down_kernel(const float* __restrict__ fused, const float* __restrict__ dw,
                                const float* __restrict__ db, const int* __restrict__ counts,
                                const int* __restrict__ list_slot, float* __restrict__ out_slot) {
  int e = blockIdx.x, mt = blockIdx.y;
  int cnt = counts[e];
  if (mt * 16 >= cnt) return;
  int lane = threadIdx.x & 31, wv = threadIdx.x >> 5;
  int rr = lane & 15, kb = (lane >> 4) << 1, hi8 = (lane >> 4) << 3;
  int p = mt * 16 + rr;
  int pc = min(p, cnt - 1);
  const float* Arow = fused + (size_t)(e * S_ + pc) * F_;
  const float* B = dw + (size_t)e * F_ * H_;
  const float* bias = db + e * H_;
  int n0 = wv * 64;
  v8f acc[4] = {};
#pragma unroll 2
  for (int kk = 0; kk < F_; kk += 4) {
    v2f a = *(const v2f*)(Arow + kk + kb);
    const float* B0 = B + (size_t)(kk + kb) * H_ + n0 + rr;
    const float* B1 = B0 + H_;
#pragma unroll
    for (int nt = 0; nt < 4; nt++) {
      v2f b; b.x = B0[nt * 16]; b.y = B1[nt * 16];
      acc[nt] = wmma4(a, b, acc[nt]);
    }
  }
#pragma unroll
  for (int r2 = 0; r2 < 8; r2++) {
    int prow = mt * 16 + r2 + hi8;
    if (prow < cnt) {
      int slot = list_slot[e * S_ + prow];
#pragma unroll
      for (int nt = 0; nt < 4; nt++) {
        int col = n0 + nt * 16 + rr;
        out_slot[(size_t)slot * H_ + col] = acc[nt][r2] + bias[col];
      }
    }
  }
}

// ---------------- Final: out = x2 + sum_x tw[x] * out_slot[t,x] ----------------
__global__ void final_kernel(const float* __restrict__ x2, const float* __restrict__ out_slot,
                             const float* __restrict__ top_w, float* __restrict__ out) {
  int t = blockIdx.x;
  for (int d = threadIdx.x; d < H_; d += blockDim.x) {
    float acc = x2[t * H_ + d];
#pragma unroll
    for (int x = 0; x < KTOP_; x++)
      acc += top_w[t * KTOP_ + x] * out_slot[(size_t)(t * KTOP_ + x) * H_ + d];
    out[t * H_ + d] = acc;
  }
}

extern "C" void kernel_launch(void* const* d_in, const int* in_sizes, int n_in,
                              void* d_out, int out_size, void* d_ws, size_t ws_size,
                              hipStream_t stream) {
  const float* hidden = (const float*)d_in[0];
  const int*   pos    = (const int*)d_in[1];
  const float* ln1    = (const float*)d_in[2];
  const float* ln2    = (const float*)d_in[3];
  const float* q_w    = (const float*)d_in[4];
  const float* q_b    = (const float*)d_in[5];
  const float* k_w    = (const float*)d_in[6];
  const float* k_b    = (const float*)d_in[7];
  const float* v_w    = (const float*)d_in[8];
  const float* v_b    = (const float*)d_in[9];
  const float* o_w    = (const float*)d_in[10];
  const float* o_b    = (const float*)d_in[11];
  const float* sinks  = (const float*)d_in[12];
  const float* rw     = (const float*)d_in[13];
  const float* rb     = (const float*)d_in[14];
  const float* gup    = (const float*)d_in[15];
  const float* gub    = (const float*)d_in[16];
  const float* dwn    = (const float*)d_in[17];
  const float* dbn    = (const float*)d_in[18];
  float* out = (float*)d_out;

  // workspace layout (floats then ints)
  float* wsf   = (float*)d_ws;
  float* h     = wsf;               // 128*512
  float* qbuf  = h     + 65536;     // 128*512
  float* kbuf  = qbuf  + 65536;     // 128*128
  float* vbuf  = kbuf  + 16384;     // 128*128
  float* attn  = vbuf  + 16384;     // 128*512
  float* x2    = attn  + 65536;     // 128*512
  float* h2    = x2    + 65536;     // 128*512
  float* topw  = h2    + 65536;     // 512
  float* oslot = topw  + 512;       // 512*512
  float* fused = oslot + 262144;    // 32*128*256
  int* wsi     = (int*)(fused + 1048576);
  int* topi    = wsi;               // 512
  int* counts  = topi   + 512;      // 32
  int* ltok    = counts + 32;       // 32*128
  int* lslot   = ltok   + 4096;     // 32*128

  rms_kernel<<<S_, 256, 0, stream>>>(hidden, ln1, h);
  qkv_kernel<<<dim3(S_ / 16, QKVN_ / 16), 32, 0, stream>>>(h, q_w, q_b, k_w, k_b, v_w, v_b,
                                                           qbuf, kbuf, vbuf);
  rope_kernel<<<(S_ * NH_ * (HD_ / 2) + 255) / 256, 256, 0, stream>>>(qbuf, pos, NH_);
  rope_kernel<<<(S_ * NKV_ * (HD_ / 2) + 255) / 256, 256, 0, stream>>>(kbuf, pos, NKV_);
  attn_kernel<<<NH_, 128, 0, stream>>>(qbuf, kbuf, vbuf, sinks, attn);
  oproj_kernel<<<dim3(S_ / 16, H_ / 16), 32, 0, stream>>>(attn, o_w, o_b, hidden, x2);
  rms_kernel<<<S_, 256, 0, stream>>>(x2, ln2, h2);
  router_kernel<<<S_, 32, 0, stream>>>(h2, rw, rb, topi, topw);
  scatter_kernel<<<1, 512, 0, stream>>>(topi, counts, ltok, lslot);
  moe_gu_kernel<<<dim3(E_, 8), 256, 0, stream>>>(h2, gup, gub, counts, ltok, fused);
  moe_down_kernel<<<dim3(E_, 8), 256, 0, stream>>>(fused, dwn, dbn, counts, lslot, oslot);
  final_kernel<<<S_, 256, 0, stream>>>(x2, oslot, topw, out);
}